// Observer_73297911873828
// MI455X (gfx1250) — compile-verified
//
#include <hip/hip_runtime.h>

// ---------------------------------------------------------------------------
// Segmented per-row min/max observer (8192x8192 f32, 64 groups, act-order map)
// MI455X / gfx1250: memory-bound (256MB @ 23.3TB/s ~= 11us floor).
// Strategy: async global->LDS tile staging (ASYNCcnt), LDS float min/max
// atomics with conflict-free lane=row layout (wave32), fused qparams epilogue.
// ---------------------------------------------------------------------------

#define NTHREADS 256
#define RPB      32          // rows per block (== wave32 width: lane = row)
#define TW       64          // tile width in columns
#define TPITCH   66          // padded LDS pitch (floats): 8B aligned, 66%64=2
#define NG       64
#define ROWS_C   8192
#define COLS_C   8192
#define NTILES   (COLS_C / TW)   // 128

typedef int v2i __attribute__((ext_vector_type(2)));
typedef __attribute__((address_space(1))) v2i* gptr_v2i;
typedef __attribute__((address_space(3))) v2i* lptr_v2i;

#if defined(__has_builtin)
#  if __has_builtin(__builtin_amdgcn_global_load_async_to_lds_b64)
#    define USE_ASYNC 1
#  endif
#endif
#ifndef USE_ASYNC
#  define USE_ASYNC 0
#endif

__device__ __forceinline__ void copy_b64_to_lds(const void* gsrc, void* ldst) {
#if USE_ASYNC
  // global_load_async_to_lds_b64: per-lane global addr -> per-lane LDS addr,
  // tracked with ASYNCcnt (CDNA5 ISA).
  __builtin_amdgcn_global_load_async_to_lds_b64(
      (gptr_v2i)gsrc, (lptr_v2i)ldst, 0, 0);
#else
  *(float2*)ldst = *(const float2*)gsrc;   // global_load_b64 + ds_store_b64
#endif
}

__device__ __forceinline__ void wait_async_and_barrier() {
#if USE_ASYNC
#  if __has_builtin(__builtin_amdgcn_s_wait_asynccnt)
  __builtin_amdgcn_s_wait_asynccnt(0);
#  else
  asm volatile("s_wait_asynccnt 0x0" ::: "memory");
#  endif
#endif
  __syncthreads();
}

__global__ __launch_bounds__(NTHREADS) void obs_minmax_kernel(
    const float* __restrict__ obs, const int* __restrict__ gidx,
    float* __restrict__ out_scale, int* __restrict__ out_zp)
{
  // acc[g][row]: a wave's 32 atomic lanes hit 32 consecutive DWORDs ->
  // 32 distinct LDS banks, no same-address serialization despite random g.
  __shared__ float accMin[NG * RPB];                 // 8 KB
  __shared__ float accMax[NG * RPB];                 // 8 KB
  __shared__ float tile[2][RPB * TPITCH];            // ~16.5 KB, double buffer
  __shared__ int   gtile[2][TW];                     // 0.5 KB

  const int tid  = threadIdx.x;
  const int wave = tid >> 5;
  const int lane = tid & 31;        // lane == local row
  const int row0 = blockIdx.x * RPB;

  for (int i = tid; i < NG * RPB; i += NTHREADS) {
    accMin[i] =  __builtin_inff();
    accMax[i] = -__builtin_inff();
  }

  // ---- prologue: stage tile 0 into buffer 0 (coalesced b64 per lane) ----
  {
    #pragma unroll
    for (int rep = 0; rep < 4; ++rep) {
      const int q  = tid + rep * NTHREADS;  // 0..1023 -> 32 rows x 32 float2
      const int rl = q >> 5;
      const int c2 = q & 31;
      copy_b64_to_lds(obs + (size_t)(row0 + rl) * COLS_C + c2 * 2,
                      &tile[0][rl * TPITCH + c2 * 2]);
    }
    if (tid < 32)
      copy_b64_to_lds(gidx + tid * 2, &gtile[0][tid * 2]);
  }

  for (int t = 0; t < NTILES; ++t) {
    wait_async_and_barrier();           // tile t resident; buffer t^1 free
    const int cur = t & 1;

    if (t + 1 < NTILES) {               // prefetch tile t+1 into other buffer
      const int nb    = cur ^ 1;
      const int cbase = (t + 1) * TW;
      #pragma unroll
      for (int rep = 0; rep < 4; ++rep) {
        const int q  = tid + rep * NTHREADS;
        const int rl = q >> 5;
        const int c2 = q & 31;
        copy_b64_to_lds(obs + (size_t)(row0 + rl) * COLS_C + cbase + c2 * 2,
                        &tile[nb][rl * TPITCH + c2 * 2]);
      }
      if (tid < 32)
        copy_b64_to_lds(gidx + cbase + tid * 2, &gtile[nb][tid * 2]);
    }

    // ---- process tile t: wave owns 8 columns, lane = row ----
    // tile read banks: (lane*66 + c) % 64 = 2*lane + c -> distinct per lane.
    const float* trow = &tile[cur][lane * TPITCH + wave * 8];
    const int*   gp   = &gtile[cur][wave * 8];
    #pragma unroll
    for (int c = 0; c < 8; ++c) {
      const float v = trow[c];
      const int   g = gp[c];            // wave-uniform -> LDS broadcast read
      __hip_atomic_fetch_min(&accMin[g * RPB + lane], v,
                             __ATOMIC_RELAXED, __HIP_MEMORY_SCOPE_WORKGROUP);
      __hip_atomic_fetch_max(&accMax[g * RPB + lane], v,
                             __ATOMIC_RELAXED, __HIP_MEMORY_SCOPE_WORKGROUP);
    }
  }

  __syncthreads();

  // ---- epilogue: qparams, coalesced writes (i = rl*64 + g) ----
  for (int i = tid; i < RPB * NG; i += NTHREADS) {
    const int rl = i >> 6;
    const int g  = i & (NG - 1);
    float mn = fminf(accMin[g * RPB + rl], 0.0f);   // include zero in range
    float mx = fmaxf(accMax[g * RPB + rl], 0.0f);   // empty group -> 0/0
    float scale = (mx - mn) / 15.0f;                // QMAX-QMIN = 15
    scale = fmaxf(scale, 1.1920929e-07f);           // EPS clamp
    float zpf = rintf(-8.0f - mn / scale);          // RNE like jnp.round
    zpf = fminf(fmaxf(zpf, -8.0f), 7.0f);
    const int oi = (row0 + rl) * NG + g;
    out_scale[oi] = scale;
    out_zp[oi]    = (int)zpf;
  }
}

extern "C" void kernel_launch(void* const* d_in, const int* in_sizes, int n_in,
                              void* d_out, int out_size, void* d_ws, size_t ws_size,
                              hipStream_t stream) {
  (void)in_sizes; (void)n_in; (void)out_size; (void)d_ws; (void)ws_size;
  const float* obs = (const float*)d_in[0];
  const int*   gix = (const int*)d_in[1];
  float* out_scale = (float*)d_out;                       // 8192*64 f32
  int*   out_zp    = (int*)d_out + (size_t)ROWS_C * NG;   // 8192*64 i32
  hipLaunchKernelGGL(obs_minmax_kernel, dim3(ROWS_C / RPB), dim3(NTHREADS),
                     0, stream, obs, gix, out_scale, out_zp);
}